// SparselyGatedMoE_81046032876006
// MI455X (gfx1250) — compile-verified
//
#include <hip/hip_runtime.h>
#include <math.h>

typedef float v2f __attribute__((ext_vector_type(2)));
typedef float v8f __attribute__((ext_vector_type(8)));

#define DDIM 1024
#define HDIM 4096
#define NEXP 8
#define NTOK 8192
#define HC   128            // H chunk per iteration (8 waves x 16 cols)
#define XLD  (DDIM + 4)     // padded LDS row stride (floats) -> conflict-free
#define HLD  (HC + 4)
#define WLD  68             // staged weight tile row stride (64 + 4 floats)
#define WLDB (WLD * 4)      // 272 bytes
#define TILEF (16 * WLD)    // floats per 16x64 wave tile (1088)
#define BUFF  (8 * TILEF)   // floats per buffer (8 waves)            (8704)
#define BUFB  (BUFF * 4)    // bytes per buffer                       (34816)

// ---- CDNA5 async memory->LDS helpers (inline asm; portable across toolchains)
__device__ __forceinline__ unsigned ldsAddrOf(const void* p) {
    // generic LDS address: low 32 bits are the LDS byte offset (ISA 10.2)
    return (unsigned)(unsigned long long)p;
}
__device__ __forceinline__ void asyncLoadB128(unsigned ldsAddr, const void* gptr) {
    asm volatile("global_load_async_to_lds_b128 %0, %1, off"
                 :: "v"(ldsAddr), "v"(gptr) : "memory");
}
__device__ __forceinline__ void waitAsync8() {   // older of 2 in-flight chunks done
    asm volatile("s_wait_asynccnt 8" ::: "memory");
}
__device__ __forceinline__ void waitAsync0() {
    asm volatile("s_wait_asynccnt 0" ::: "memory");
}
// stage a 16-row x 64-col fp32 tile: 8 async b128, 2 rows per instruction
__device__ __forceinline__ void stage16x64(unsigned lds0, const float* g0, size_t rowStrideF) {
    #pragma unroll
    for (int i = 0; i < 8; ++i)
        asyncLoadB128(lds0 + (unsigned)(i * 2 * WLDB), g0 + (size_t)(2 * i) * rowStrideF);
}

// ---------------------------------------------------------------------------
// Kernel 1: noisy top-2 gating + per-expert compaction
// ---------------------------------------------------------------------------
__global__ __launch_bounds__(256)
void moe_gate_kernel(const float* __restrict__ x, const float* __restrict__ noise,
                     const float* __restrict__ gate_w, const float* __restrict__ gate_b,
                     const float* __restrict__ var_w, const float* __restrict__ var_b,
                     int* __restrict__ cnt, int* __restrict__ tokList,
                     int* __restrict__ slotList, float* __restrict__ coefList) {
    __shared__ __align__(16) float gw[NEXP * DDIM];
    __shared__ __align__(16) float vw[NEXP * DDIM];
    const int tid = threadIdx.x;
    for (int i = tid; i < NEXP * DDIM / 4; i += 256) {
        ((float4*)gw)[i] = ((const float4*)gate_w)[i];
        ((float4*)vw)[i] = ((const float4*)var_w)[i];
    }
    __syncthreads();

    const int wave = tid >> 5, lane = tid & 31;
    const int t = blockIdx.x * 8 + wave;

    float xr[32];
    const float* xrow = x + (size_t)t * DDIM;
    #pragma unroll
    for (int i = 0; i < 32; ++i) xr[i] = xrow[lane + 32 * i];

    float noisy[NEXP];
    for (int e = 0; e < NEXP; ++e) {
        float ag = 0.f, av = 0.f;
        #pragma unroll
        for (int i = 0; i < 32; ++i) {
            const int d = lane + 32 * i;
            ag = fmaf(xr[i], gw[e * DDIM + d], ag);
            av = fmaf(xr[i], vw[e * DDIM + d], av);
        }
        #pragma unroll
        for (int off = 16; off; off >>= 1) {
            ag += __shfl_xor(ag, off, 32);
            av += __shfl_xor(av, off, 32);
        }
        const float logit = ag + gate_b[e];
        const float v = av + var_b[e];
        const float stdv = fmaxf(v, 0.f) + log1pf(expf(-fabsf(v)));  // stable softplus
        noisy[e] = logit + noise[(size_t)t * NEXP + e] * stdv;
    }

    int i1 = 0; float n1 = noisy[0];
    #pragma unroll
    for (int e = 1; e < NEXP; ++e) if (noisy[e] > n1) { n1 = noisy[e]; i1 = e; }
    int i2 = -1; float n2 = -INFINITY;
    #pragma unroll
    for (int e = 0; e < NEXP; ++e) if (e != i1 && noisy[e] > n2) { n2 = noisy[e]; i2 = e; }
    const float c1 = 1.f / (1.f + expf(n2 - n1));
    const float c2 = 1.f - c1;

    if (lane == 0) {
        int p = atomicAdd(&cnt[i1], 1);
        tokList[i1 * NTOK + p] = t; slotList[i1 * NTOK + p] = 0; coefList[i1 * NTOK + p] = c1;
        p = atomicAdd(&cnt[i2], 1);
        tokList[i2 * NTOK + p] = t; slotList[i2 * NTOK + p] = 1; coefList[i2 * NTOK + p] = c2;
    }
}

// ---------------------------------------------------------------------------
// Kernel 2: fused expert FFN; fp32 WMMA 16x16x4,
//           double-buffered async-to-LDS weight streaming
// ---------------------------------------------------------------------------
__global__ __launch_bounds__(256)
void moe_ffn_kernel(const float* __restrict__ x,
                    const float* __restrict__ w1, const float* __restrict__ b1,
                    const float* __restrict__ w2, const float* __restrict__ b2,
                    const int* __restrict__ cnt, const int* __restrict__ tokList,
                    const int* __restrict__ slotList, const float* __restrict__ coefList,
                    float* __restrict__ yslot) {
    const int e = blockIdx.x;
    const int tileBase = blockIdx.y * 16;
    const int ne = cnt[e];
    if (tileBase >= ne) return;

    // weight staging first -> low LDS offsets
    __shared__ __align__(16) float W1s[2 * BUFF];    // 2 x (8 waves x 16 x WLD)
    __shared__ __align__(16) float W2s[2 * BUFF];
    __shared__ __align__(16) float Xs[16 * XLD];     // token tile
    __shared__ __align__(16) float Hs[16 * HLD];     // relu(h) chunk
    __shared__ float coefS[16];
    __shared__ int   tokS[16];
    __shared__ int   rowS[16];

    const int tid = threadIdx.x, wave = tid >> 5, lane = tid & 31;
    const int hi = lane >> 4;          // half-wave (K offset 0/2, staging row parity)
    const int lm = lane & 15;          // M (A) or N (B,C) index
    const int koff = 2 * hi;
    const int hc4 = lm * 4;            // staging column (floats)

    if (tid < 16) {
        const int r = tileBase + tid;
        const bool valid = (r < ne);
        const int tok = valid ? tokList[e * NTOK + r] : 0;
        tokS[tid]  = tok;
        rowS[tid]  = valid ? (tok * 2 + slotList[e * NTOK + r]) : -1;
        coefS[tid] = valid ? coefList[e * NTOK + r] : 0.f;
    }
    __syncthreads();

    // stage X tile (coalesced float4; zero-fill padding rows)
    for (int rr = wave * 2; rr < wave * 2 + 2; ++rr) {
        float4* dst = (float4*)(Xs + rr * XLD);
        if (rowS[rr] >= 0) {
            const float4* src = (const float4*)(x + (size_t)tokS[rr] * DDIM);
            for (int i = lane; i < DDIM / 4; i += 32) dst[i] = src[i];
        } else {
            const float4 z = {0.f, 0.f, 0.f, 0.f};
            for (int i = lane; i < DDIM / 4; i += 32) dst[i] = z;
        }
    }
    __syncthreads();

    const float* w1e = w1 + (size_t)e * HDIM * DDIM;
    const float* w2e = w2 + (size_t)e * DDIM * HDIM;

    // per-wave, per-buffer staging bases (byte addrs) and read bases
    const unsigned laneOff = (unsigned)(hi * WLDB + lm * 16);
    const unsigned w1L0 = ldsAddrOf(W1s) + (unsigned)(wave * TILEF * 4) + laneOff;
    const unsigned w1L1 = w1L0 + BUFB;
    const unsigned w2L0 = ldsAddrOf(W2s) + (unsigned)(wave * TILEF * 4) + laneOff;
    const unsigned w2L1 = w2L0 + BUFB;
    const float* w1T0 = W1s + wave * TILEF + lm * WLD;    // B-operand row base
    const float* w1T1 = w1T0 + BUFF;
    const float* w2T0 = W2s + wave * TILEF + lm * WLD;
    const float* w2T1 = w2T0 + BUFF;

    v8f yacc[8];
    #pragma unroll
    for (int j = 0; j < 8; ++j) yacc[j] = (v8f){0.f,0.f,0.f,0.f,0.f,0.f,0.f,0.f};

    for (int it = 0; it < HDIM / HC; ++it) {
        const int hcBase = it * HC;

        // ======== GEMM1: this wave's 16 h-cols, K = D as 16 chunks of 64 ========
        v8f c1 = (v8f){0.f,0.f,0.f,0.f,0.f,0.f,0.f,0.f};
        auto g1c = [&](int c, const float* tile) {           // 16 WMMAs per chunk
            #pragma unroll
            for (int kk = 0; kk < 64; kk += 4) {
                v2f a = *(const v2f*)(Xs + lm * XLD + c * 64 + kk + koff);
                v2f b = *(const v2f*)(tile + kk + koff);
                c1 = __builtin_amdgcn_wmma_f32_16x16x4_f32(false, a, false, b,
                                                           (short)0, c1, false, false);
            }
        };
        {
            const float* g1 = w1e + (size_t)(hcBase + wave * 16 + hi) * DDIM + hc4;
            stage16x64(w1L0, g1, DDIM);                           // chunk 0 -> buf0
            #pragma unroll 1
            for (int c = 0; c + 2 < 16; c += 2) {
                stage16x64(w1L1, g1 + (c + 1) * 64, DDIM);        // c+1 -> buf1
                waitAsync8();  g1c(c, w1T0);
                stage16x64(w1L0, g1 + (c + 2) * 64, DDIM);        // c+2 -> buf0
                waitAsync8();  g1c(c + 1, w1T1);
            }
            stage16x64(w1L1, g1 + 15 * 64, DDIM);                 // chunk 15 -> buf1
            waitAsync8();  g1c(14, w1T0);
            waitAsync0();  g1c(15, w1T1);
        }
        const float b1v = b1[(size_t)e * HDIM + hcBase + wave * 16 + lm];

        __syncthreads();   // prior GEMM2 readers of Hs are done
        #pragma unroll
        for (int v = 0; v < 8; ++v) {
            float hval = c1[v] + b1v;                             // C row m = v + 8*hi
            Hs[(v + 8 * hi) * HLD + wave * 16 + lm] = fmaxf(hval, 0.f);
        }
        __syncthreads();

        // ======== GEMM2: 8 d-tiles x 2 half-chunks, pipelined across j ========
        {
            const float* g2 = w2e + (size_t)(wave * 128 + hi) * HDIM + hcBase + hc4;
            stage16x64(w2L0, g2, HDIM);                           // (j=0, half=0) -> buf0
            #pragma unroll
            for (int j = 0; j < 8; ++j) {
                const float* gj = g2 + (size_t)j * 16 * HDIM;
                stage16x64(w2L1, gj + 64, HDIM);                  // (j,1) -> buf1
                waitAsync8();
                v8f acc = yacc[j];
                #pragma unroll
                for (int kk = 0; kk < 64; kk += 4) {              // (j,0) from buf0
                    v2f a = *(const v2f*)(Hs + lm * HLD + kk + koff);
                    v2f b = *(const v2f*)(w2T0 + kk + koff);
                    acc = __builtin_amdgcn_wmma_f32_16x16x4_f32(false, a, false, b,
                                                                (short)0, acc, false, false);
                }
                if (j < 7) { stage16x64(w2L0, gj + 16 * HDIM, HDIM);  // (j+1,0) -> buf0
                             waitAsync8(); }
                else       { waitAsync0(); }
                #pragma unroll
                for (int kk = 0; kk < 64; kk += 4) {              // (j,1) from buf1
                    v2f a = *(const v2f*)(Hs + lm * HLD + 64 + kk + koff);
                    v2f b = *(const v2f*)(w2T1 + kk + koff);
                    acc = __builtin_amdgcn_wmma_f32_16x16x4_f32(false, a, false, b,
                                                                (short)0, acc, false, false);
                }
                yacc[j] = acc;
            }
        }
    }

    // finalize: y = (acc + b2) * coef -> per-(token,slot) buffer (deterministic)
    #pragma unroll
    for (int j = 0; j < 8; ++j) {
        const int dcol = wave * 128 + j * 16 + lm;
        const float b2v = b2[(size_t)e * DDIM + dcol];
        #pragma unroll
        for (int v = 0; v < 8; ++v) {
            const int m = v + 8 * hi;
            const int row = rowS[m];
            if (row >= 0)
                yslot[(size_t)row * DDIM + dcol] = (yacc[j][v] + b2v) * coefS[m];
        }
    }
}

// ---------------------------------------------------------------------------
// Kernel 3: combine the two expert slots per token
// ---------------------------------------------------------------------------
__global__ __launch_bounds__(256)
void moe_combine_kernel(const float* __restrict__ ys, float* __restrict__ out) {
    const size_t idx = (size_t)blockIdx.x * 256 + threadIdx.x;   // over N*D/4
    const size_t q = DDIM / 4;
    const size_t t = idx / q, c = idx % q;
    const float4 a = ((const float4*)ys)[(t * 2 + 0) * q + c];
    const float4 b = ((const float4*)ys)[(t * 2 + 1) * q + c];
    float4 o;
    o.x = a.x + b.x; o.y = a.y + b.y; o.z = a.z + b.z; o.w = a.w + b.w;
    ((float4*)out)[idx] = o;
}

// ---------------------------------------------------------------------------
extern "C" void kernel_launch(void* const* d_in, const int* in_sizes, int n_in,
                              void* d_out, int out_size, void* d_ws, size_t ws_size,
                              hipStream_t stream) {
    const float* x      = (const float*)d_in[0];
    const float* noise  = (const float*)d_in[1];
    const float* gate_w = (const float*)d_in[2];
    const float* gate_b = (const float*)d_in[3];
    const float* var_w  = (const float*)d_in[4];
    const float* var_b  = (const float*)d_in[5];
    const float* w1     = (const float*)d_in[6];
    const float* b1     = (const float*)d_in[7];
    const float* w2     = (const float*)d_in[8];
    const float* b2     = (const float*)d_in[9];
    float* out = (float*)d_out;

    char* ws = (char*)d_ws;
    int*   cnt      = (int*)ws;                                  // 32 B (pad 256)
    int*   tokList  = (int*)(ws + 256);                          // E*N ints
    int*   slotList = tokList + NEXP * NTOK;
    float* coefList = (float*)(slotList + NEXP * NTOK);
    float* yslot    = (float*)(ws + 256 + 3 * (size_t)NEXP * NTOK * 4); // N*2*D floats

    hipMemsetAsync(cnt, 0, NEXP * sizeof(int), stream);

    moe_gate_kernel<<<NTOK / 8, 256, 0, stream>>>(
        x, noise, gate_w, gate_b, var_w, var_b, cnt, tokList, slotList, coefList);

    moe_ffn_kernel<<<dim3(NEXP, NTOK / 16), 256, 0, stream>>>(
        x, w1, b1, w2, b2, cnt, tokList, slotList, coefList, yslot);

    moe_combine_kernel<<<(NTOK * DDIM / 4) / 256, 256, 0, stream>>>(yslot, out);
}